// GCBlock_65532611002568
// MI455X (gfx1250) — compile-verified
//
#include <hip/hip_runtime.h>
#include <hip/hip_bf16.h>
#include <math.h>

#define NN 50000
#define EE 800000
#define DN 128
#define DM 64
#define RR 50
#define HH 4
#define CC 16
#define PAD 2   // LDS row padding (floats) -> 8B-aligned rows for v2f reads

typedef __attribute__((ext_vector_type(2))) float v2f;
typedef __attribute__((ext_vector_type(8))) float v8f;

// D = A(16x4 f32) * B(4x16 f32) + C  (V_WMMA_F32_16X16X4_F32)
__device__ inline v8f wmma4(v2f a, v2f b, v8f c) {
    return __builtin_amdgcn_wmma_f32_16x16x4_f32(
        false, a, false, b, (short)0, c, false, false);
}

// order-preserving float <-> uint mapping for atomicMax on floats
__device__ inline unsigned f2o(float f) {
    unsigned u = __float_as_uint(f);
    return (u & 0x80000000u) ? ~u : (u | 0x80000000u);
}
__device__ inline float o2f(unsigned u) {
    u = (u & 0x80000000u) ? (u & 0x7FFFFFFFu) : ~u;
    return __uint_as_float(u);
}

__device__ inline v2f ld2(const float* p) { return *(const v2f*)p; }

// ---------------------------------------------------------------------------
// Kernel 0: W_comb[R,DM] = W_edge[R,DM] @ W_e[DM,DM]   (tiny, scalar)
// ---------------------------------------------------------------------------
__global__ void wcomb_kernel(const float* __restrict__ We,
                             const float* __restrict__ Wl,
                             float* __restrict__ Wc) {
    int idx = blockIdx.x * blockDim.x + threadIdx.x;
    if (idx >= RR * DM) return;
    int r = idx / DM, n = idx % DM;
    float acc = 0.f;
    for (int j = 0; j < DM; ++j) acc += We[r * DM + j] * Wl[j * DM + n];
    Wc[idx] = acc;
}

// ---------------------------------------------------------------------------
// Kernel 1: init accumulators (amax = -inf key, den = 0, agg = 0)
// ---------------------------------------------------------------------------
__global__ void init_kernel(unsigned* __restrict__ amax,
                            float* __restrict__ den,
                            float* __restrict__ agg) {
    int i = blockIdx.x * blockDim.x + threadIdx.x;
    if (i < NN * HH) { amax[i] = 0u; den[i] = 0.f; }
    if (i < NN * DM) agg[i] = 0.f;
}

// ---------------------------------------------------------------------------
// Kernel 2: per 16-node tile: xm = x@W_n2m (WMMA K=128), then through LDS
//           q/k/v/x_r = xm@W + b (WMMA K=64).  One wave per tile.
// ---------------------------------------------------------------------------
__global__ __launch_bounds__(32) void node_linear_kernel(
    const float* __restrict__ x, const float* __restrict__ Wn2m,
    const float* __restrict__ Wq, const float* __restrict__ bq,
    const float* __restrict__ Wk, const float* __restrict__ bk,
    const float* __restrict__ Wv, const float* __restrict__ bv,
    const float* __restrict__ Wskip, const float* __restrict__ bskip,
    float* __restrict__ qo, float* __restrict__ ko,
    float* __restrict__ vo, float* __restrict__ xro) {
    __shared__ float xm_s[16][DM + PAD];
    const int lane = threadIdx.x;
    const int row16 = lane & 15;
    const int kh = lane >> 4;            // 0: K={0,1}, 1: K={2,3}
    const int base = blockIdx.x * 16;    // exact: 50000/16 = 3125 tiles
    const float* xrow = x + (size_t)(base + row16) * DN;

    // xm tile (16 x 64), 4 column tiles, K = 128
    for (int t = 0; t < 4; ++t) {
        v8f acc = {};
        const int ncol = t * 16 + row16;
        for (int kk = 0; kk < DN; kk += 4) {
            const int k0 = kk + 2 * kh;                 // even -> 8B aligned
            v2f a = ld2(xrow + k0);                     // global_load_b64
            v2f b;
            b.x = Wn2m[k0 * DM + ncol];
            b.y = Wn2m[(k0 + 1) * DM + ncol];
            acc = wmma4(a, b, acc);
        }
#pragma unroll
        for (int r = 0; r < 8; ++r) xm_s[r + 8 * kh][ncol] = acc[r];
    }
    __syncthreads();

    const float* Ws[4] = {Wq, Wk, Wv, Wskip};
    const float* bs[4] = {bq, bk, bv, bskip};
    float* os[4] = {qo, ko, vo, xro};

    for (int o = 0; o < 4; ++o) {
        const float* __restrict__ W = Ws[o];
        const float* __restrict__ bb = bs[o];
        float* __restrict__ out = os[o];
        for (int t = 0; t < 4; ++t) {
            v8f acc = {};
            const int ncol = t * 16 + row16;
            for (int kk = 0; kk < DM; kk += 4) {
                const int k0 = kk + 2 * kh;
                v2f a = ld2(&xm_s[row16][k0]);          // ds_load_b64
                v2f b;
                b.x = W[k0 * DM + ncol];
                b.y = W[(k0 + 1) * DM + ncol];
                acc = wmma4(a, b, acc);
            }
            const float bias = bb[ncol];
#pragma unroll
            for (int r = 0; r < 8; ++r)
                out[(size_t)(base + r + 8 * kh) * DM + ncol] = acc[r] + bias;
        }
    }
}

// ---------------------------------------------------------------------------
// Kernel 3: ee[E,64] = edge_attr[E,50] @ W_comb[50,64]   (WMMA, K padded 52)
// ---------------------------------------------------------------------------
__global__ __launch_bounds__(32) void edge_ee_kernel(
    const float* __restrict__ ea, const float* __restrict__ Wc,
    float* __restrict__ ee) {
    const int lane = threadIdx.x;
    const int row16 = lane & 15;
    const int kh = lane >> 4;
    const size_t base = (size_t)blockIdx.x * 16;  // exact: 800000/16 tiles
    const float* earow = ea + (base + row16) * RR;

    for (int t = 0; t < 4; ++t) {
        v8f acc = {};
        const int ncol = t * 16 + row16;
        for (int kk = 0; kk < RR; kk += 4) {      // kk = 0..48; K guarded to 50
            const int k0 = kk + 2 * kh;
            // RR is even -> the {k0, k0+1} pair is valid or invalid as a unit
            v2f a = {0.f, 0.f}, b = {0.f, 0.f};
            if (k0 + 1 < RR) {
                a = ld2(earow + k0);              // 8B aligned (200B row, even k0)
                b.x = Wc[k0 * DM + ncol];
                b.y = Wc[(k0 + 1) * DM + ncol];
            }
            acc = wmma4(a, b, acc);
        }
#pragma unroll
        for (int r = 0; r < 8; ++r)
            ee[(base + r + 8 * kh) * DM + ncol] = acc[r];
    }
}

// ---------------------------------------------------------------------------
// Kernel 4: per-edge attention logits + running per-(dst,head) max
// ---------------------------------------------------------------------------
__global__ void alpha_max_kernel(const int* __restrict__ ei,
                                 const float* __restrict__ q,
                                 const float* __restrict__ k,
                                 const float* __restrict__ ee,
                                 float* __restrict__ alpha,
                                 unsigned* __restrict__ amax) {
    const int e = blockIdx.x * blockDim.x + threadIdx.x;
    if (e >= EE) return;
    const int src = ei[e];
    const int dst = ei[EE + e];
    const float4* qv = (const float4*)(q + (size_t)dst * DM);
    const float4* kv = (const float4*)(k + (size_t)src * DM);
    const float4* ev = (const float4*)(ee + (size_t)e * DM);
#pragma unroll
    for (int h = 0; h < HH; ++h) {
        float s = 0.f;
#pragma unroll
        for (int c4 = 0; c4 < CC / 4; ++c4) {
            float4 qq = qv[h * 4 + c4];
            float4 kk = kv[h * 4 + c4];
            float4 ex = ev[h * 4 + c4];
            s += qq.x * (kk.x + ex.x) + qq.y * (kk.y + ex.y) +
                 qq.z * (kk.z + ex.z) + qq.w * (kk.w + ex.w);
        }
        s *= 0.25f;  // 1/sqrt(C), C=16
        alpha[(size_t)e * HH + h] = s;
        atomicMax(&amax[dst * HH + h], f2o(s));
    }
}

// ---------------------------------------------------------------------------
// Kernel 5: exp, denominator, and UNNORMALIZED weighted scatter-add
// ---------------------------------------------------------------------------
__global__ void scatter_kernel(const int* __restrict__ ei,
                               const float* __restrict__ v,
                               const float* __restrict__ ee,
                               const float* __restrict__ alpha,
                               const unsigned* __restrict__ amax,
                               float* __restrict__ den,
                               float* __restrict__ agg) {
    const int e = blockIdx.x * blockDim.x + threadIdx.x;
    if (e >= EE) return;
    const int src = ei[e];
    const int dst = ei[EE + e];
#pragma unroll
    for (int h = 0; h < HH; ++h) {
        const float m = o2f(amax[dst * HH + h]);
        const float ex = __expf(alpha[(size_t)e * HH + h] - m);
        atomicAdd(&den[dst * HH + h], ex);
#pragma unroll
        for (int c = 0; c < CC; ++c) {
            const int ch = h * CC + c;
            const float val = v[(size_t)src * DM + ch] + ee[(size_t)e * DM + ch];
            atomicAdd(&agg[(size_t)dst * DM + ch], ex * val);
        }
    }
}

// ---------------------------------------------------------------------------
// Kernel 6: normalize, beta gate, final GEMM out[16,64]@W_m2n[64,128] (WMMA)
// ---------------------------------------------------------------------------
__global__ __launch_bounds__(32) void finalize_kernel(
    const float* __restrict__ agg, const float* __restrict__ den,
    const float* __restrict__ xr, const float* __restrict__ Wbeta,
    const float* __restrict__ Wm2n, float* __restrict__ out) {
    __shared__ float s_o[16][DM + PAD];
    __shared__ float s_r[16][DM + PAD];
    __shared__ float s_beta[16];
    const int lane = threadIdx.x;
    const int row16 = lane & 15;
    const int kh = lane >> 4;
    const int base = blockIdx.x * 16;

    for (int idx = lane; idx < 16 * DM; idx += 32) {
        const int row = idx >> 6, col = idx & 63;
        const int node = base + row;
        const float d = den[node * HH + (col >> 4)];
        float o = agg[(size_t)node * DM + col];
        o = (d > 0.f) ? (o / d) : 0.f;   // isolated node -> 0 (matches ref)
        s_o[row][col] = o;
        s_r[row][col] = xr[(size_t)node * DM + col];
    }
    __syncthreads();

    if (lane < 16) {
        float acc = 0.f;
        for (int c = 0; c < DM; ++c) {
            const float o = s_o[lane][c], r = s_r[lane][c];
            acc += o * Wbeta[c] + r * Wbeta[DM + c] + (o - r) * Wbeta[2 * DM + c];
        }
        s_beta[lane] = 1.f / (1.f + __expf(-acc));
    }
    __syncthreads();

    for (int idx = lane; idx < 16 * DM; idx += 32) {
        const int row = idx >> 6, col = idx & 63;
        const float b = s_beta[row];
        s_o[row][col] = b * s_r[row][col] + (1.f - b) * s_o[row][col];
    }
    __syncthreads();

    for (int t = 0; t < 8; ++t) {        // 8 column tiles of 16 -> D_NODE=128
        v8f acc = {};
        const int ncol = t * 16 + row16;
        for (int kk = 0; kk < DM; kk += 4) {
            const int k0 = kk + 2 * kh;
            v2f a = ld2(&s_o[row16][k0]);               // ds_load_b64
            v2f b;
            b.x = Wm2n[k0 * DN + ncol];
            b.y = Wm2n[(k0 + 1) * DN + ncol];
            acc = wmma4(a, b, acc);
        }
#pragma unroll
        for (int r = 0; r < 8; ++r)
            out[(size_t)(base + r + 8 * kh) * DN + ncol] = acc[r];
    }
}

// ---------------------------------------------------------------------------
extern "C" void kernel_launch(void* const* d_in, const int* in_sizes, int n_in,
                              void* d_out, int out_size, void* d_ws, size_t ws_size,
                              hipStream_t stream) {
    const float* x      = (const float*)d_in[0];
    const float* eattr  = (const float*)d_in[1];
    const int*   eidx   = (const int*)  d_in[2];
    // d_in[3] = batch (unused)
    const float* Wn2m   = (const float*)d_in[4];
    const float* Wedge  = (const float*)d_in[5];
    const float* Wq     = (const float*)d_in[6];
    const float* bq     = (const float*)d_in[7];
    const float* Wk     = (const float*)d_in[8];
    const float* bk     = (const float*)d_in[9];
    const float* Wv     = (const float*)d_in[10];
    const float* bv     = (const float*)d_in[11];
    const float* We     = (const float*)d_in[12];
    const float* Wskip  = (const float*)d_in[13];
    const float* bskip  = (const float*)d_in[14];
    const float* Wbeta  = (const float*)d_in[15];
    const float* Wm2n   = (const float*)d_in[16];
    float* out = (float*)d_out;

    // workspace layout (floats)
    float* ws   = (float*)d_ws;
    float* q    = ws;                              // N*DM
    float* k    = q   + (size_t)NN * DM;           // N*DM
    float* v    = k   + (size_t)NN * DM;           // N*DM
    float* xr   = v   + (size_t)NN * DM;           // N*DM
    float* agg  = xr  + (size_t)NN * DM;           // N*DM
    float* den  = agg + (size_t)NN * DM;           // N*H
    unsigned* amax = (unsigned*)(den + (size_t)NN * HH);  // N*H
    float* alpha = (float*)(amax + (size_t)NN * HH);      // E*H
    float* Wc   = alpha + (size_t)EE * HH;         // R*DM
    float* ee   = Wc + (size_t)RR * DM;            // E*DM

    // 0) fold W_edge @ W_e -> W_comb  (kills the [E,64] intermediate)
    wcomb_kernel<<<(RR * DM + 255) / 256, 256, 0, stream>>>(Wedge, We, Wc);
    // 1) init accumulators
    init_kernel<<<(NN * DM + 255) / 256, 256, 0, stream>>>(amax, den, agg);
    // 2) node-side fused linears (WMMA f32)
    node_linear_kernel<<<NN / 16, 32, 0, stream>>>(
        x, Wn2m, Wq, bq, Wk, bk, Wv, bv, Wskip, bskip, q, k, v, xr);
    // 3) ee = edge_attr @ W_comb (WMMA f32)
    edge_ee_kernel<<<EE / 16, 32, 0, stream>>>(eattr, Wc, ee);
    // 4) logits + segment max
    alpha_max_kernel<<<(EE + 255) / 256, 256, 0, stream>>>(eidx, q, k, ee, alpha, amax);
    // 5) exp + denominator + unnormalized scatter
    scatter_kernel<<<(EE + 255) / 256, 256, 0, stream>>>(eidx, v, ee, alpha, amax, den, agg);
    // 6) normalize + beta gate + final WMMA GEMM
    finalize_kernel<<<NN / 16, 32, 0, stream>>>(agg, den, xr, Wbeta, Wm2n, out);
}